// CapsuleLayer_63282048139564
// MI455X (gfx1250) — compile-verified
//
#include <hip/hip_runtime.h>
#include <hip/hip_bf16.h>
#include <math.h>

// ---------------------------------------------------------------------------
// CapsNet forward for MI455X (gfx1250, wave32, WMMA).
// Conv = implicit GEMM via V_WMMA_F32_16X16X4_F32 (fp32-exact, compute bound).
// Routing = bandwidth-trivial VALU kernels, kernel-boundary global sync.
// ---------------------------------------------------------------------------

typedef float v2f __attribute__((ext_vector_type(2)));
typedef float v8f __attribute__((ext_vector_type(8)));

static __device__ __forceinline__ v8f wmma_f32(v2f a, v2f b, v8f c) {
    // D = A(16x4 f32) * B(4x16 f32) + C(16x16 f32)
    return __builtin_amdgcn_wmma_f32_16x16x4_f32(
        /*neg_a=*/false, a, /*neg_b=*/false, b,
        /*c_mod=*/(short)0, c, /*reuse_a=*/false, /*reuse_b=*/false);
}

// ---------------------------------------------------------------------------
// Kernel 1: grouped conv 9x9 stride2 as implicit GEMM.
//   Out[p, ch] = sum_{ci,kh,kw} In[b, ci, 2*oh+kh, 2*ow+kw] * W[ch, ci, kh, kw]
//   p = (b, oh, ow): M = 8 imgs * 36 = 288 per WG (18 M-tiles of 16)
//   ch tile: 32 per WG (2 N-tiles of 16)
//   K ordered (ci, khw): per ci stage image slices + weight rows in LDS,
//   khw padded 81 -> 84 (pad rows of B are zero) => 21 WMMA K-steps.
// Writes x_raw[b][r][k] with r = o*36 + oh*6 + ow, ch = k*32 + o.
// ---------------------------------------------------------------------------
#define IMG_PER_WG 8
#define CH_PER_WG  32

__global__ __launch_bounds__(288) void conv_caps_wmma(
    const float* __restrict__ In,    // [128][256][20][20]
    const float* __restrict__ Wg,    // [256][256][9][9] flat = [256][20736]
    const float* __restrict__ Bias,  // [256]  (flat [8][32], ch = k*32+o)
    float* __restrict__ Xraw)        // [128][1152][8]
{
    __shared__ float img[IMG_PER_WG * 400];   // 8 raw 20x20 slices for this ci
    __shared__ float Bt[84 * CH_PER_WG];      // B tile [khw(padded)][ch_local]

    const int tid     = threadIdx.x;
    const int ch_base = blockIdx.x * CH_PER_WG;   // 0..224
    const int b_base  = blockIdx.y * IMG_PER_WG;  // 0..120
    const int lane  = tid & 31;
    const int wave  = tid >> 5;        // 0..8
    const int mrow  = lane & 15;       // A-fragment row within tile
    const int khalf = lane >> 4;       // 0/1: which K half this lane holds

    // Zero the padded B rows (81..83) once; never overwritten afterwards.
    if (tid < 3 * CH_PER_WG) Bt[81 * CH_PER_WG + tid] = 0.0f;

    // Per-lane A base addresses for the wave's two M-tiles.
    const int pA0 = wave * 16 + mrow;          // 0..143
    const int pA1 = pA0 + 144;                 // 144..287
    const int base0 = (pA0 / 36) * 400 + ((pA0 % 36) / 6) * 40 + (pA0 % 6) * 2;
    const int base1 = (pA1 / 36) * 400 + ((pA1 % 36) / 6) * 40 + (pA1 % 6) * 2;
    // Per-lane B columns for the two N-tiles.
    const int col0 = mrow;
    const int col1 = mrow + 16;

    v8f acc00 = {}; v8f acc01 = {}; v8f acc10 = {}; v8f acc11 = {};

    for (int ci = 0; ci < 256; ++ci) {
        // ---- stage: 8 image slices (coalesced 400-float rows) ----
        for (int idx = tid; idx < IMG_PER_WG * 400; idx += 288) {
            int bi = idx / 400, j = idx % 400;
            img[idx] = In[((size_t)(b_base + bi) * 256 + ci) * 400 + j];
        }
        // ---- stage: 32 weight rows of 81, transposed into Bt[khw][ch] ----
        for (int idx = tid; idx < CH_PER_WG * 81; idx += 288) {
            int nn = idx / 81, khw = idx % 81;
            Bt[khw * CH_PER_WG + nn] =
                Wg[(size_t)(ch_base + nn) * 20736 + (size_t)ci * 81 + khw];
        }
        __syncthreads();

        // ---- 21 K-steps of WMMA (K = 4 each, khw padded to 84) ----
#pragma unroll
        for (int kk = 0; kk < 21; ++kk) {
            const int c0 = kk * 4, c1 = c0 + 1, c2 = c0 + 2, c3 = c0 + 3;
            // compile-time khw -> image offset (kh*20 + kw); pads map to 0
            const int o0 = (c0 < 81) ? ((c0 / 9) * 20 + c0 % 9) : 0;
            const int o1 = (c1 < 81) ? ((c1 / 9) * 20 + c1 % 9) : 0;
            const int o2 = (c2 < 81) ? ((c2 / 9) * 20 + c2 % 9) : 0;
            const int o3 = (c3 < 81) ? ((c3 / 9) * 20 + c3 % 9) : 0;
            const int offA0 = khalf ? o2 : o0;
            const int offA1 = khalf ? o3 : o1;

            v2f a0, a1, b0, b1;
            a0.x = img[base0 + offA0]; a0.y = img[base0 + offA1];
            a1.x = img[base1 + offA0]; a1.y = img[base1 + offA1];
            const int kb = c0 + khalf * 2;
            b0.x = Bt[kb * CH_PER_WG + col0]; b0.y = Bt[(kb + 1) * CH_PER_WG + col0];
            b1.x = Bt[kb * CH_PER_WG + col1]; b1.y = Bt[(kb + 1) * CH_PER_WG + col1];

            acc00 = wmma_f32(a0, b0, acc00);
            acc01 = wmma_f32(a0, b1, acc01);
            acc10 = wmma_f32(a1, b0, acc10);
            acc11 = wmma_f32(a1, b1, acc11);
        }
        __syncthreads();
    }

    // ---- epilogue: bias add + scatter to x_raw[b][r][k] ----
#pragma unroll
    for (int mt = 0; mt < 2; ++mt) {
#pragma unroll
        for (int nt = 0; nt < 2; ++nt) {
            v8f acc = (mt == 0) ? ((nt == 0) ? acc00 : acc01)
                                : ((nt == 0) ? acc10 : acc11);
            const int ch = ch_base + nt * 16 + mrow;  // D-matrix N = lane%16
            const float bias = Bias[ch];
            const int k = ch >> 5, o = ch & 31;
#pragma unroll
            for (int j = 0; j < 8; ++j) {
                // D layout: VGPR j -> M = j (lanes 0-15) / j+8 (lanes 16-31)
                const int pm = mt * 144 + wave * 16 + j + khalf * 8;
                const int b = b_base + pm / 36;
                const int q = pm % 36;
                const int r = o * 36 + q;
                Xraw[((size_t)b * 1152 + r) * 8 + k] = acc[j] + bias;
            }
        }
    }
}

// ---------------------------------------------------------------------------
// Kernel 2: squash x over the capsule-group dim (8), in place.
// ---------------------------------------------------------------------------
__global__ __launch_bounds__(256) void squash_primary(float* __restrict__ X)
{
    const int i = blockIdx.x * blockDim.x + threadIdx.x;
    if (i >= 128 * 1152) return;
    float* p = X + (size_t)i * 8;
    float t[8];
    float sn = 0.0f;
#pragma unroll
    for (int j = 0; j < 8; ++j) { t[j] = p[j]; sn += t[j] * t[j]; }
    const float f = sn / ((1.0f + sn) * sqrtf(sn));
#pragma unroll
    for (int j = 0; j < 8; ++j) p[j] = t[j] * f;
}

// ---------------------------------------------------------------------------
// Kernel 3: pred[c,b,r,o] = sum_i x[b,r,i] * RW[c,r,i,o]
// WG = (rblock of 16, c); LDS-tiled so RW is read once from HBM.
// ---------------------------------------------------------------------------
__global__ __launch_bounds__(256) void pred_kernel(
    const float* __restrict__ X,   // [128][1152][8]
    const float* __restrict__ RW,  // [10][1152][8][16]
    float* __restrict__ Pred)      // [10][128][1152][16]
{
    __shared__ float Wl[16 * 128];    // [rr][i][o]
    __shared__ float Xl[128 * 128];   // [b][rr][i]
    const int c  = blockIdx.y;
    const int r0 = blockIdx.x * 16;
    const int t  = threadIdx.x;

    for (int idx = t; idx < 2048; idx += 256)
        Wl[idx] = RW[(size_t)c * 147456 + (size_t)r0 * 128 + idx];
    for (int idx = t; idx < 16384; idx += 256) {
        const int b = idx >> 7, rem = idx & 127;
        Xl[idx] = X[(size_t)b * 9216 + (size_t)r0 * 8 + rem];
    }
    __syncthreads();

    for (int m = t; m < 32768; m += 256) {
        const int b = m >> 8, rr = (m >> 4) & 15, o = m & 15;
        const float* xp = &Xl[b * 128 + rr * 8];
        const float* wp = &Wl[rr * 128 + o];
        float s = 0.0f;
#pragma unroll
        for (int i = 0; i < 8; ++i) s += xp[i] * wp[i * 16];
        Pred[(((size_t)c * 128 + b) * 1152 + (r0 + rr)) * 16 + o] = s;
    }
}

// ---------------------------------------------------------------------------
// Kernel 4: s[c,b,:] = sum_r softmax_r(logits[c,b,:])_r * pred[c,b,r,:]
// iter 0 uses uniform weights (logits are all-zero in the reference).
// ---------------------------------------------------------------------------
__global__ __launch_bounds__(256) void routing_s(
    const float* __restrict__ Pred, const float* __restrict__ Logits,
    float* __restrict__ S, int use_logits)
{
    const int b = blockIdx.x, c = blockIdx.y, t = threadIdx.x;
    __shared__ float red[256];
    __shared__ float redv[16 * 256];

    const float* lg = Logits + ((size_t)c * 128 + b) * 1152;
    float mx = 0.0f, inv_denom = 1.0f / 1152.0f;

    if (use_logits) {
        float m = -1e30f;
        for (int r = t; r < 1152; r += 256) m = fmaxf(m, lg[r]);
        red[t] = m; __syncthreads();
        for (int s2 = 128; s2 > 0; s2 >>= 1) {
            if (t < s2) red[t] = fmaxf(red[t], red[t + s2]);
            __syncthreads();
        }
        mx = red[0]; __syncthreads();
        float sum = 0.0f;
        for (int r = t; r < 1152; r += 256) sum += __expf(lg[r] - mx);
        red[t] = sum; __syncthreads();
        for (int s2 = 128; s2 > 0; s2 >>= 1) {
            if (t < s2) red[t] += red[t + s2];
            __syncthreads();
        }
        inv_denom = 1.0f / red[0]; __syncthreads();
    }

    float a[16];
#pragma unroll
    for (int o = 0; o < 16; ++o) a[o] = 0.0f;
    const float* pp = Pred + ((size_t)c * 128 + b) * 1152 * 16;
    for (int r = t; r < 1152; r += 256) {
        const float w = use_logits ? __expf(lg[r] - mx) * inv_denom : inv_denom;
        const float* row = pp + (size_t)r * 16;
#pragma unroll
        for (int o = 0; o < 16; ++o) a[o] += w * row[o];
    }
#pragma unroll
    for (int o = 0; o < 16; ++o) redv[o * 256 + t] = a[o];
    for (int s2 = 128; s2 > 0; s2 >>= 1) {
        __syncthreads();
        for (int e = t; e < 16 * s2; e += 256) {
            const int o = e / s2, tt = e - o * s2;
            redv[o * 256 + tt] += redv[o * 256 + tt + s2];
        }
    }
    __syncthreads();
    if (t < 16) S[((size_t)c * 128 + b) * 16 + t] = redv[t * 256];
}

// ---------------------------------------------------------------------------
// Kernel 5: v = squash(s, axis=batch)  (faithful to the reference quirk)
//   sn[c,o] = sum_b s[c,b,o]^2 ; v[c,b,o] = s * sn/((1+sn)*sqrt(sn))
// ---------------------------------------------------------------------------
__global__ __launch_bounds__(256) void routing_v(
    const float* __restrict__ S, float* __restrict__ V,
    float* __restrict__ Out, int last)
{
    const int c = blockIdx.x, t = threadIdx.x;
    __shared__ float red[256];
    __shared__ float fac[16];
    const float* sp = S + (size_t)c * 2048;

    const int o = t & 15, g = t >> 4;   // 16 groups over batch
    float p = 0.0f;
    for (int b = g; b < 128; b += 16) { const float v = sp[b * 16 + o]; p += v * v; }
    red[t] = p; __syncthreads();
    for (int s2 = 8; s2 > 0; s2 >>= 1) {
        if (g < s2) red[t] += red[t + s2 * 16];
        __syncthreads();
    }
    if (t < 16) {
        const float sn = red[t];
        fac[t] = sn / ((1.0f + sn) * sqrtf(sn));
    }
    __syncthreads();

    for (int idx = t; idx < 2048; idx += 256) {
        const int b = idx >> 4, oo = idx & 15;
        const float v = sp[idx] * fac[oo];
        if (last) Out[((size_t)b * 10 + c) * 16 + oo] = v;   // [B,10,16]
        else      V[(size_t)c * 2048 + idx] = v;
    }
}

// ---------------------------------------------------------------------------
// Kernel 6: logits[c,b,r] (+)= sum_o pred[c,b,r,o] * v[c,b,o]
// iter 0 writes (logits buffer needs no zero-init), iter 1 accumulates.
// ---------------------------------------------------------------------------
__global__ __launch_bounds__(256) void routing_logits(
    const float* __restrict__ Pred, const float* __restrict__ V,
    float* __restrict__ Logits, int accum)
{
    const int b = blockIdx.x, c = blockIdx.y, t = threadIdx.x;
    __shared__ float vl[16];
    if (t < 16) vl[t] = V[(size_t)c * 2048 + b * 16 + t];
    __syncthreads();
    const float* pp = Pred + ((size_t)c * 128 + b) * 1152 * 16;
    float* lg = Logits + ((size_t)c * 128 + b) * 1152;
    for (int r = t; r < 1152; r += 256) {
        const float* row = pp + (size_t)r * 16;
        float d = 0.0f;
#pragma unroll
        for (int oo = 0; oo < 16; ++oo) d += row[oo] * vl[oo];
        lg[r] = accum ? (lg[r] + d) : d;
    }
}

// ---------------------------------------------------------------------------
extern "C" void kernel_launch(void* const* d_in, const int* in_sizes, int n_in,
                              void* d_out, int out_size, void* d_ws, size_t ws_size,
                              hipStream_t stream)
{
    const float* In   = (const float*)d_in[0];  // [128,256,20,20]
    const float* Wg   = (const float*)d_in[1];  // [8,32,256,9,9]
    const float* Bias = (const float*)d_in[2];  // [8,32]
    const float* RW   = (const float*)d_in[3];  // [10,1152,8,16]
    float* Out = (float*)d_out;                 // [128,10,16]

    // workspace layout (floats): ~105.2 MB total
    float* X      = (float*)d_ws;               // 128*1152*8   = 1,179,648
    float* Pred   = X + 1179648;                // 10*128*1152*16 = 23,592,960
    float* Logits = Pred + 23592960;            // 10*128*1152  = 1,474,560
    float* S      = Logits + 1474560;           // 10*128*16    = 20,480
    float* V      = S + 20480;                  // 10*128*16    = 20,480

    // 1) conv as implicit GEMM (fp32 WMMA): grid = 8 ch-groups x 16 img-groups
    conv_caps_wmma<<<dim3(8, 16), 288, 0, stream>>>(In, Wg, Bias, X);
    // 2) primary-capsule squash (in place)
    squash_primary<<<(128 * 1152 + 255) / 256, 256, 0, stream>>>(X);
    // 3) prediction vectors
    pred_kernel<<<dim3(72, 10), 256, 0, stream>>>(X, RW, Pred);
    // 4) dynamic routing, 3 iterations (kernel boundaries = global sync)
    for (int it = 0; it < 3; ++it) {
        routing_s<<<dim3(128, 10), 256, 0, stream>>>(Pred, Logits, S, it > 0 ? 1 : 0);
        routing_v<<<10, 256, 0, stream>>>(S, V, Out, it == 2 ? 1 : 0);
        if (it < 2)
            routing_logits<<<dim3(128, 10), 256, 0, stream>>>(Pred, V, Logits, it > 0 ? 1 : 0);
    }
}